// Model_20409684591278
// MI455X (gfx1250) — compile-verified
//
#include <hip/hip_runtime.h>
#include <hip/hip_bf16.h>
#include <math.h>
#include <stdint.h>

typedef __attribute__((ext_vector_type(2))) float v2f;
typedef __attribute__((ext_vector_type(8))) float v8f;

#define NS  51   // seq positions
#define NV  64   // vocab
#define NE  4    // embedding dim
#define NH1 5
#define NH2 7
#define ND  64   // output dim
#define SLICE_DW (NV * ND)   // 4096 dwords = 16KB per position slice of T

__device__ __forceinline__ float leaky(float a) { return a > 0.0f ? a : 0.01f * a; }

__device__ __forceinline__ float fast_tanh(float x) {
#if __has_builtin(__builtin_amdgcn_tanhf)
  return __builtin_amdgcn_tanhf(x);           // gfx1250 v_tanh_f32 (TRANS unit)
#else
  float t = __expf(-2.0f * __builtin_fabsf(x));
  float r = __fdividef(1.0f - t, 1.0f + t);
  return __builtin_copysignf(r, x);
#endif
}

// ---------------------------------------------------------------------------
// Kernel 1: T[s][v][d] = tanh( W3[s] @ h2(s,v) + b3[s] )   (3264 x 64 table)
// One wave per position s; vocab 64 rows via 4 WMMA M-tiles, D=64 via 4 N-tiles.
// ---------------------------------------------------------------------------
__global__ __launch_bounds__(32) void build_table(
    const float* __restrict__ emb,
    const float* __restrict__ W1, const float* __restrict__ b1,
    const float* __restrict__ W2, const float* __restrict__ b2,
    const float* __restrict__ W3, const float* __restrict__ b3,
    float* __restrict__ T) {
  const int s    = blockIdx.x;
  const int lane = threadIdx.x;
  const int nloc = lane & 15;
  const int hsel = lane >> 4;

  // h2 for vocab entries v = lane and v = lane + 32 (pure VALU, tiny)
  float h2[2][8];
  #pragma unroll
  for (int half = 0; half < 2; ++half) {
    const int v = half * 32 + lane;
    const float4 e = *(const float4*)(emb + ((s * NV + v) << 2));
    float h1[NH1];
    #pragma unroll
    for (int i = 0; i < NH1; ++i) {
      const float4 w = *(const float4*)(W1 + ((s * NH1 + i) << 2));
      h1[i] = leaky(b1[s * NH1 + i] + w.x * e.x + w.y * e.y + w.z * e.z + w.w * e.w);
    }
    #pragma unroll
    for (int i = 0; i < NH2; ++i) {
      const float* wr = W2 + (s * NH2 + i) * NH1;
      float a = b2[s * NH2 + i];
      #pragma unroll
      for (int j = 0; j < NH1; ++j) a += wr[j] * h1[j];
      h2[half][i] = leaky(a);
    }
    h2[half][7] = 0.0f;  // K pad 7 -> 8
  }

  // B fragments from W3[s] (f32 4x16 layout: lane L vgpr v -> K=2*(L/16)+v, N=L%16)
  const float* W3s = W3 + s * (ND * NH2);
  float bfr[4][2][2];
  #pragma unroll
  for (int t = 0; t < 4; ++t) {
    const float* wrow = W3s + (t * 16 + nloc) * NH2;
    #pragma unroll
    for (int k = 0; k < 2; ++k) {
      int g0 = 4 * k + 2 * hsel;
      int g1 = g0 + 1;
      if (g1 == NH2) g1 = 0;  // padded K=7 column: matching A column is zero
      bfr[t][k][0] = wrow[g0];
      bfr[t][k][1] = wrow[g1];
    }
  }

  // 4 M-tiles over vocab; A fragments via cross-lane shuffle of h2
  #pragma unroll
  for (int m = 0; m < 4; ++m) {
    const int srcLane = (m & 1) * 16 + nloc;
    float afr[2][2];
    #pragma unroll
    for (int k = 0; k < 2; ++k) {
      #pragma unroll
      for (int v = 0; v < 2; ++v) {
        float lo = __shfl(h2[m >> 1][4 * k + v],     srcLane, 32);
        float hi = __shfl(h2[m >> 1][4 * k + 2 + v], srcLane, 32);
        afr[k][v] = hsel ? hi : lo;
      }
    }
    v2f A0 = {afr[0][0], afr[0][1]};
    v2f A1 = {afr[1][0], afr[1][1]};
    #pragma unroll
    for (int t = 0; t < 4; ++t) {
      v2f B0 = {bfr[t][0][0], bfr[t][0][1]};
      v2f B1 = {bfr[t][1][0], bfr[t][1][1]};
      float c = b3[s * ND + t * 16 + nloc];
      v8f acc = {c, c, c, c, c, c, c, c};
      acc = __builtin_amdgcn_wmma_f32_16x16x4_f32(false, A0, false, B0,
                                                  (short)0, acc, false, false);
      acc = __builtin_amdgcn_wmma_f32_16x16x4_f32(false, A1, false, B1,
                                                  (short)0, acc, false, false);
      #pragma unroll
      for (int j = 0; j < 8; ++j) {
        const int row = m * 16 + hsel * 8 + j;             // vocab index
        T[(s * NV + row) * ND + t * 16 + nloc] = fast_tanh(acc[j]);
      }
    }
  }
}

// ---------------------------------------------------------------------------
// Kernel 2: out[b,:] = sum_s T[s, x[b,s], :]
// Triple-buffered async global->LDS staging of the 16KB T[s] slice,
// rotation-swizzled ((cc+v)&15) LDS layout to spread banks for the random-row gather.
// ---------------------------------------------------------------------------
__device__ __forceinline__ void async_b128_to_lds(unsigned lds_addr, const void* gaddr) {
  asm volatile("global_load_async_to_lds_b128 %0, %1, off"
               :: "v"(lds_addr), "v"(gaddr) : "memory");
}

__global__ __launch_bounds__(256) void gather_sum(
    const int* __restrict__ x, const float* __restrict__ T,
    float* __restrict__ out) {
  __shared__ __align__(16) float lds_T[3 * SLICE_DW];   // 48 KB, 3 slices
  const int tid = threadIdx.x;
  const unsigned ldsBase = (unsigned)(uintptr_t)&lds_T[0];
  const int r = blockIdx.x * 256 + tid;                 // one batch row per lane

  // Stage slice `src` of T into buffer `bslot` (4 async b128 per thread).
  // Element chunk cc (16B) of vocab row v goes to slot (cc+v)&15 (bank swizzle).
  auto stage = [&](int src, int bslot) {
    const float* g = T + src * SLICE_DW;
    #pragma unroll
    for (int k = 0; k < 4; ++k) {
      const int u  = tid + 256 * k;        // 16B unit index within slice
      const int v  = u >> 4;               // vocab row
      const int cc = u & 15;               // 16B chunk within row
      const unsigned dst =
          ldsBase + ((unsigned)(bslot * SLICE_DW + (v << 6) + (((cc + v) & 15) << 2)) << 2);
      async_b128_to_lds(dst, (const void*)(g + (u << 2)));
    }
  };

  stage(0, 0);
  stage(1, 1);
  asm volatile("s_wait_asynccnt 0x4" ::: "memory");   // slice 0 resident
  __syncthreads();

  float4 acc[16];
  #pragma unroll
  for (int c = 0; c < 16; ++c) acc[c] = make_float4(0.f, 0.f, 0.f, 0.f);

  const int* xrow = x + r * NS;

  for (int s = 0; s < NS; ++s) {
    // Prefetch slice s+2 into rotating buffer (dummy re-stage at tail keeps the
    // wait discipline uniform; that buffer is never read again).
    const int nxt = s + 2;
    stage(nxt < NS ? nxt : 0, nxt % 3);

    if (s + 8 < NS) __builtin_prefetch(xrow + s + 8, 0, 1);  // x stream prefetch
    const int idx = xrow[s] & (NV - 1);

    const float* slice = &lds_T[(s % 3) * SLICE_DW];
    const int base = idx << 6;
    #pragma unroll
    for (int c = 0; c < 16; ++c) {
      const float4 vv = *(const float4*)&slice[base + (((c + idx) & 15) << 2)];
      acc[c].x += vv.x; acc[c].y += vv.y; acc[c].z += vv.z; acc[c].w += vv.w;
    }

    // batch for slice s+1 must be done (only the just-issued 4 may remain);
    // our ds reads must be complete before anyone overwrites this buffer.
    asm volatile("s_wait_asynccnt 0x4\n\ts_wait_dscnt 0x0" ::: "memory");
    __syncthreads();
  }

  float* orow = out + (size_t)r * ND;
  #pragma unroll
  for (int c = 0; c < 16; ++c) *(float4*)&orow[c * 4] = acc[c];
}

// ---------------------------------------------------------------------------
extern "C" void kernel_launch(void* const* d_in, const int* in_sizes, int n_in,
                              void* d_out, int out_size, void* d_ws, size_t ws_size,
                              hipStream_t stream) {
  const int*   x   = (const int*)d_in[0];
  const float* emb = (const float*)d_in[1];
  const float* W1  = (const float*)d_in[2];
  const float* b1  = (const float*)d_in[3];
  const float* W2  = (const float*)d_in[4];
  const float* b2  = (const float*)d_in[5];
  const float* W3  = (const float*)d_in[6];
  const float* b3  = (const float*)d_in[7];
  float* out = (float*)d_out;
  float* T   = (float*)d_ws;               // 51*64*64*4 = 835,584 bytes

  build_table<<<dim3(NS), dim3(32), 0, stream>>>(emb, W1, b1, W2, b2, W3, b3, T);

  const int B = in_sizes[0] / NS;          // 65536
  gather_sum<<<dim3(B / 256), dim3(256), 0, stream>>>(x, T, out);
}